// GCN3_bias_20727512170664
// MI455X (gfx1250) — compile-verified
//
#include <hip/hip_runtime.h>

typedef __attribute__((ext_vector_type(2))) float v2f;
typedef __attribute__((ext_vector_type(8))) float v8f;

#define FDIM 64

// ---------------- zero the 3 scalar accumulators ----------------
__global__ void zero3_kernel(float* ws) {
    if (threadIdx.x < 3) ws[threadIdx.x] = 0.0f;
}

// ---------------- out[row,f] = emb[row,f] * d[row]  (float4) ----------------
__global__ void init_scale_kernel(float* __restrict__ out, const float* __restrict__ emb,
                                  const float* __restrict__ d, long n4) {
    long i = (long)blockIdx.x * blockDim.x + threadIdx.x;
    if (i >= n4) return;
    long row = (i * 4) >> 6;                 // FDIM = 64
    float dv = d[row];
    float4 e = ((const float4*)emb)[i];
    float4 o = make_float4(e.x * dv, e.y * dv, e.z * dv, e.w * dv);
    ((float4*)out)[i] = o;
}

// ---------------- out[rows[e],:] += vals[e] * x[cols[e],:]  ----------------
// 16 threads per edge, 4 floats per thread: b128 gather + 4 coalesced f32 atomics
__global__ void scatter_kernel(float* __restrict__ out, const float* __restrict__ x,
                               const float* __restrict__ vals,
                               const int* __restrict__ rows, const int* __restrict__ cols,
                               int E) {
    long t = (long)blockIdx.x * blockDim.x + threadIdx.x;
    if (t >= (long)E * 16) return;
    int e = (int)(t >> 4);
    int f = (int)(t & 15) * 4;
    float val = vals[e];
    int c = cols[e];
    int r = rows[e];
    float4 xv = *(const float4*)(x + (size_t)c * FDIM + f);
    float* o = out + (size_t)r * FDIM + f;
    atomicAdd(o + 0, val * xv.x);
    atomicAdd(o + 1, val * xv.y);
    atomicAdd(o + 2, val * xv.z);
    atomicAdd(o + 3, val * xv.w);
}

// ---------------- in-place relu (float4) ----------------
__global__ void relu_kernel(float* __restrict__ buf, long n4) {
    long i = (long)blockIdx.x * blockDim.x + threadIdx.x;
    if (i >= n4) return;
    float4 v = ((float4*)buf)[i];
    v.x = fmaxf(v.x, 0.0f); v.y = fmaxf(v.y, 0.0f);
    v.z = fmaxf(v.z, 0.0f); v.w = fmaxf(v.w, 0.0f);
    ((float4*)buf)[i] = v;
}

// ---------------- dst = w0*e0 + w1*g1 + w2*dst (in place), accumulate sum(dst^2) ----------------
__global__ void combine_kernel(float* __restrict__ dst, const float* __restrict__ e0,
                               const float* __restrict__ g1, const float* __restrict__ add_w,
                               float* __restrict__ ss, long n) {
    long i = (long)blockIdx.x * blockDim.x + threadIdx.x;
    float w0 = add_w[0], w1 = add_w[1], w2 = add_w[2];
    float v = 0.0f;
    if (i < n) {
        v = w0 * e0[i] + w1 * g1[i] + w2 * dst[i];
        dst[i] = v;
    }
    float sq = v * v;
    // wave32 reduction
    for (int off = 16; off > 0; off >>= 1) sq += __shfl_down(sq, off, 32);
    __shared__ float partial[8];
    int lane = threadIdx.x & 31;
    int wid  = threadIdx.x >> 5;
    if (lane == 0) partial[wid] = sq;
    __syncthreads();
    if (threadIdx.x == 0) {
        float t = 0.0f;
        #pragma unroll
        for (int w = 0; w < 8; ++w) t += partial[w];
        atomicAdd(ss, t);
    }
}

// ---------------- batched dot products via chained V_WMMA_F32_16X16X4_F32 ----------------
// One wave handles 16 predictions. A[m][k] = u_m[k]*it_m[k] (16x4 per step),
// B = ones(4x16)  =>  C[m][n] = partial dot for row m in every column n.
// C layout (f32 16x16): lanes 0-15 hold M=r (VGPR r), lanes 16-31 hold M=r+8.
// So lane 0 reads preds 0..7 from acc[0..7]; lane 16 reads preds 8..15.
__global__ void pred_wmma_kernel(const float* __restrict__ gu, const float* __restrict__ gi,
                                 const int* __restrict__ user0, const int* __restrict__ item0,
                                 const float* __restrict__ user_bias,
                                 const float* __restrict__ item_bias,
                                 const float* __restrict__ avg_rating,
                                 const float* __restrict__ ratings,
                                 float* __restrict__ sse, int B) {
    int lane = threadIdx.x & 31;
    long wave = ((long)blockIdx.x * blockDim.x + threadIdx.x) >> 5;
    long base = wave * 16;
    if (base >= B) return;                    // uniform per wave -> EXEC stays all-ones

    int row   = lane & 15;                    // which prediction row this lane feeds
    int khalf = (lane >> 4) * 2;              // A-matrix f32 16x4 layout: hi lanes carry K=2,3
    int uidx = user0[base + row];
    int iidx = item0[base + row];
    const float* up = gu + (size_t)uidx * FDIM;
    const float* ip = gi + (size_t)iidx * FDIM;

    v8f acc = {0.f, 0.f, 0.f, 0.f, 0.f, 0.f, 0.f, 0.f};
    v2f bones; bones.x = 1.0f; bones.y = 1.0f;

    #pragma unroll
    for (int c = 0; c < 16; ++c) {
        int k = c * 4 + khalf;
        v2f a;
        a.x = up[k]     * ip[k];
        a.y = up[k + 1] * ip[k + 1];
        acc = __builtin_amdgcn_wmma_f32_16x16x4_f32(
            /*neg_a=*/false, a, /*neg_b=*/false, bones,
            /*c_mod=*/(short)0, acc, /*reuse_a=*/false, /*reuse_b=*/false);
    }

    if ((lane & 15) == 0) {                   // lanes 0 and 16
        int roff = (lane >> 4) * 8;
        float avg = avg_rating[0];
        float local = 0.0f;
        #pragma unroll
        for (int r = 0; r < 8; ++r) {
            long idx = base + roff + r;
            float p = acc[r] + user_bias[user0[idx]] + item_bias[item0[idx]] + avg;
            float d = p - ratings[idx];
            local += d * d;
        }
        atomicAdd(sse, local);
    }
}

// ---------------- final 3 scalars ----------------
__global__ void finalize_kernel(const float* __restrict__ ws, float* __restrict__ out,
                                float invB, float invUF, float invIF) {
    float loss2 = ws[0] * invB;
    float l2 = 0.001f * (ws[1] * invUF) + 0.001f * (ws[2] * invIF);
    out[0] = loss2 + l2;
    out[1] = loss2;
    out[2] = l2;
}

static inline unsigned cdivu(long a, long b) { return (unsigned)((a + b - 1) / b); }

extern "C" void kernel_launch(void* const* d_in, const int* in_sizes, int n_in,
                              void* d_out, int out_size, void* d_ws, size_t ws_size,
                              hipStream_t stream) {
    const float* embed_user = (const float*)d_in[0];
    const float* embed_item = (const float*)d_in[1];
    const float* ui_vals    = (const float*)d_in[2];
    const float* iu_vals    = (const float*)d_in[3];
    const float* d_i        = (const float*)d_in[4];
    const float* d_j        = (const float*)d_in[5];
    const float* add_w      = (const float*)d_in[6];
    const float* user_bias  = (const float*)d_in[7];
    const float* item_bias  = (const float*)d_in[8];
    const float* avg_rating = (const float*)d_in[9];
    const float* ratings    = (const float*)d_in[10];
    const int*   ui_rows    = (const int*)d_in[11];
    const int*   ui_cols    = (const int*)d_in[12];
    const int*   user0      = (const int*)d_in[13];
    const int*   item_i0    = (const int*)d_in[14];

    const long U = in_sizes[0] / FDIM;
    const long I = in_sizes[1] / FDIM;
    const int  E = in_sizes[2];
    const int  B = in_sizes[10];
    const long nU = U * FDIM, nI = I * FDIM;

    float* ws  = (float*)d_ws;
    float* sse = ws + 0;
    float* ssu = ws + 1;
    float* ssi = ws + 2;
    float* g1u = ws + 64;           // 256-byte offset keeps float4 alignment
    float* g1i = g1u + nU;
    float* g2u = g1i + nI;
    float* g2i = g2u + nU;

    const int T = 256;
    float* out = (float*)d_out;

    hipLaunchKernelGGL(zero3_kernel, dim3(1), dim3(32), 0, stream, ws);

    // ---- layer 1: g1 = relu(spmm + emb*d) ----
    hipLaunchKernelGGL(init_scale_kernel, dim3(cdivu(nU / 4, T)), dim3(T), 0, stream,
                       g1u, embed_user, d_i, nU / 4);
    hipLaunchKernelGGL(init_scale_kernel, dim3(cdivu(nI / 4, T)), dim3(T), 0, stream,
                       g1i, embed_item, d_j, nI / 4);
    hipLaunchKernelGGL(scatter_kernel, dim3(cdivu((long)E * 16, T)), dim3(T), 0, stream,
                       g1u, embed_item, ui_vals, ui_rows, ui_cols, E);
    hipLaunchKernelGGL(scatter_kernel, dim3(cdivu((long)E * 16, T)), dim3(T), 0, stream,
                       g1i, embed_user, iu_vals, ui_cols, ui_rows, E);
    hipLaunchKernelGGL(relu_kernel, dim3(cdivu(nU / 4, T)), dim3(T), 0, stream, g1u, nU / 4);
    hipLaunchKernelGGL(relu_kernel, dim3(cdivu(nI / 4, T)), dim3(T), 0, stream, g1i, nI / 4);

    // ---- layer 2: g2 = relu(spmm(g1) + g1*d) ----
    hipLaunchKernelGGL(init_scale_kernel, dim3(cdivu(nU / 4, T)), dim3(T), 0, stream,
                       g2u, g1u, d_i, nU / 4);
    hipLaunchKernelGGL(init_scale_kernel, dim3(cdivu(nI / 4, T)), dim3(T), 0, stream,
                       g2i, g1i, d_j, nI / 4);
    hipLaunchKernelGGL(scatter_kernel, dim3(cdivu((long)E * 16, T)), dim3(T), 0, stream,
                       g2u, g1i, ui_vals, ui_rows, ui_cols, E);
    hipLaunchKernelGGL(scatter_kernel, dim3(cdivu((long)E * 16, T)), dim3(T), 0, stream,
                       g2i, g1u, iu_vals, ui_cols, ui_rows, E);
    hipLaunchKernelGGL(relu_kernel, dim3(cdivu(nU / 4, T)), dim3(T), 0, stream, g2u, nU / 4);
    hipLaunchKernelGGL(relu_kernel, dim3(cdivu(nI / 4, T)), dim3(T), 0, stream, g2i, nI / 4);

    // ---- gu/gi (overwrite g2 in place) + sum of squares ----
    hipLaunchKernelGGL(combine_kernel, dim3(cdivu(nU, T)), dim3(T), 0, stream,
                       g2u, embed_user, g1u, add_w, ssu, nU);
    hipLaunchKernelGGL(combine_kernel, dim3(cdivu(nI, T)), dim3(T), 0, stream,
                       g2i, embed_item, g1i, add_w, ssi, nI);

    // ---- predictions via WMMA (one wave per 16 predictions) ----
    long waves = (B + 15) / 16;
    hipLaunchKernelGGL(pred_wmma_kernel, dim3(cdivu(waves * 32, T)), dim3(T), 0, stream,
                       g2u, g2i, user0, item_i0, user_bias, item_bias,
                       avg_rating, ratings, sse, B);

    hipLaunchKernelGGL(finalize_kernel, dim3(1), dim3(1), 0, stream,
                       ws, out, 1.0f / (float)B, 1.0f / (float)nU, 1.0f / (float)nI);
}